// ARMAPosteriorModel_54554674594037
// MI455X (gfx1250) — compile-verified
//
#include <hip/hip_runtime.h>
#include <math.h>

#define S_N 256
#define T_N 1024
#define K_N 64
#define D_N 4
#define P_N 8
#define NB  16
#define LOG2PI_F 1.8378770664093453f

typedef __attribute__((ext_vector_type(2))) float v2f;
typedef __attribute__((ext_vector_type(4))) float v4f;
typedef __attribute__((ext_vector_type(8))) float v8f;

// workspace layout (float offsets)
#define WS_L     0          // [16][4][64][64] = 262144 floats (1 MB)
#define WS_C     262144     // [16][4][64]     = 4096
#define WS_SP    266240     // [T][D][P]       = 32768  (softplus(s_raw), dim_idx applied)
#define WS_LS    299008     // [T][D][P]       = 32768  (log of the above)
#define WS_AG    331776     // [T][D]          = 4096   (aG[t,d] = sigmoid(a_raw[t-1, di[d]]), aG[0]=0)
#define WS_CARRY 335872     // [16][S][D][P]   = 131072 (carry-in per block)
#define WS_W     466944     // [16][S][D][P]   = 131072 (within-block end values)
// total = 598016 floats (~2.3 MB)

// ---------------- kernel 1: elementwise precompute -----------------
__global__ void kprep(const float* __restrict__ s_raw, const float* __restrict__ a_raw,
                      const int* __restrict__ dim_idx, float* __restrict__ ws) {
  int idx = blockIdx.x * blockDim.x + threadIdx.x;
  if (idx >= T_N * D_N * P_N) return;
  int p = idx & 7;
  int d = (idx >> 3) & 3;
  int t = idx >> 5;
  int dd = dim_idx[d];
  float x = s_raw[(t * D_N + dd) * P_N + p];
  float spv = (x > 20.0f) ? x : log1pf(__expf(x));
  ws[WS_SP + idx] = spv;
  ws[WS_LS + idx] = __logf(spv);
  if (idx < T_N * D_N) {
    int td = idx >> 2, dc = idx & 3;
    int ddc = dim_idx[dc];
    float ag = 0.0f;
    if (td > 0) ag = 1.0f / (1.0f + __expf(-a_raw[(td - 1) * D_N + ddc]));
    ws[WS_AG + idx] = ag;  // multiplier applied at step td
  }
}

// ---------------- kernel 2: triangular scan matrices ---------------
// L[b,d][t'][j] = prod_{u=j+1..t'} aG[64b+u]  (unit diag, 0 above)
// c[b,d][t']    = prod_{u=0..t'}   aG[64b+u]
__global__ void kcoef(float* __restrict__ ws) {
  int b = blockIdx.x >> 2, d = blockIdx.x & 3;
  int tp = threadIdx.x;
  __shared__ float ag[K_N];
  ag[tp] = ws[WS_AG + (b * K_N + tp) * D_N + d];
  __syncthreads();
  float* Lrow = ws + WS_L + ((size_t)(b * D_N + d) * K_N + tp) * K_N;
  float prod = 1.0f;
  Lrow[tp] = 1.0f;
  for (int j = tp - 1; j >= 0; --j) { prod *= ag[j + 1]; Lrow[j] = prod; }
  for (int j = tp + 1; j < K_N; ++j) Lrow[j] = 0.0f;
  ws[WS_C + (b * D_N + d) * K_N + tp] = ag[0] * prod;
}

// ---------------- kernel 3: block-end dot products -----------------
// w[b][s][d][p] = sum_j L[b,d][63][j] * z[s, 64b+j, d, p]
__global__ void kw(const float* __restrict__ m, const float* __restrict__ noise,
                   const int* __restrict__ dim_idx, float* __restrict__ ws) {
  int idx = blockIdx.x * blockDim.x + threadIdx.x;  // NB*S*32 = 131072
  if (idx >= NB * S_N * 32) return;
  int rem = idx & 31;
  int p = rem & 7, d = rem >> 3;
  int sb = idx >> 5;
  int s = sb & (S_N - 1);
  int b = sb >> 8;
  int dd = dim_idx[d];
  const float* sp = ws + WS_SP;
  const float* aG = ws + WS_AG;
  const float* Lrow = ws + WS_L + ((size_t)(b * D_N + d) * K_N + 63) * K_N;
  float acc = 0.0f;
#pragma unroll 4
  for (int j = 0; j < K_N; ++j) {
    int t = b * K_N + j;
    float mc = 1.0f - aG[t * D_N + d];
    float nz = noise[((size_t)s * T_N + t) * 32 + d * 8 + p];
    float z = mc * m[(t * D_N + dd) * P_N + p] + sp[(t * D_N + d) * P_N + p] * nz;
    acc += Lrow[j] * z;
  }
  ws[WS_W + idx] = acc;
}

// ---------------- kernel 4: block-boundary carry scan --------------
__global__ void kscan(float* __restrict__ ws) {
  int idx = blockIdx.x * blockDim.x + threadIdx.x;
  if (idx >= S_N * D_N * P_N) return;
  int rem = idx & 31;                 // d*8 + p
  int d = rem >> 3;
  int s = idx >> 5;
  const float* c = ws + WS_C;
  const float* w = ws + WS_W;
  float* carry = ws + WS_CARRY;
  float cur = 0.0f;
  carry[(size_t)s * 32 + rem] = 0.0f;  // b = 0
  for (int b = 1; b < NB; ++b) {
    float wprev = w[((size_t)(b - 1) * S_N + s) * 32 + rem];
    float ce = c[((b - 1) * D_N + d) * K_N + 63];
    cur = wprev + ce * cur;
    carry[((size_t)b * S_N + s) * 32 + rem] = cur;
  }
}

// ---------------- kernel 5: WMMA block scan + log_prob -------------
// grid: 1024 blocks x 256 thr. blockIdx = spg*64 + (b*4+d).
// Wave wv handles sample-pair spg*8+wv: 16 cols = {s0 p0..7, s1 p0..7}.
// Accumulator is seeded with c[t'] * carry so stores are final.
__global__ void __launch_bounds__(256) kmain(const float* __restrict__ m,
                                             const float* __restrict__ noise,
                                             const int* __restrict__ dim_idx,
                                             const float* __restrict__ ws,
                                             float* __restrict__ out) {
  __shared__ float Lt[K_N * K_N];
  __shared__ float ct[K_N];
  int bd = blockIdx.x & 63;
  int b = bd >> 2, d = bd & 3;
  int spg = blockIdx.x >> 6;
  const float* Lsrc = ws + WS_L + (size_t)(b * D_N + d) * K_N * K_N;
  for (int i = threadIdx.x; i < K_N * K_N; i += 256) Lt[i] = Lsrc[i];
  if (threadIdx.x < K_N) ct[threadIdx.x] = ws[WS_C + (b * D_N + d) * K_N + threadIdx.x];
  __syncthreads();

  int lane = threadIdx.x & 31, wave = threadIdx.x >> 5;
  int col = lane & 15, half = lane >> 4;
  int s = (spg * 8 + wave) * 2 + (col >> 3);
  int p = col & 7;
  int dd = dim_idx[d];

  const float* sp = ws + WS_SP;
  const float* ls = ws + WS_LS;
  const float* aG = ws + WS_AG;
  float* out_lp = out + (size_t)S_N * T_N * D_N * P_N;

  float zc[K_N];
#pragma unroll
  for (int j = 0; j < K_N; ++j) {
    int t = b * K_N + j;
    float mc = 1.0f - aG[t * D_N + d];
    int tdp = (t * D_N + d) * P_N + p;
    float nz = noise[((size_t)s * T_N + t) * 32 + d * 8 + p];
    zc[j] = mc * m[(t * D_N + dd) * P_N + p] + sp[tdp] * nz;
    if (lane < 16)
      out_lp[((size_t)s * T_N + t) * 32 + d * 8 + p] =
          -ls[tdp] - 0.5f * LOG2PI_F - 0.5f * nz * nz;
  }

  // Seed accumulators with carry contribution: D[t'] starts at c[t'] * carry.
  float cv = ws[WS_CARRY + ((size_t)b * S_N + s) * 32 + d * 8 + p];
  v8f acc0, acc1, acc2, acc3;
#pragma unroll
  for (int r = 0; r < 8; ++r) {
    acc0[r] = ct[ 0 + half * 8 + r] * cv;
    acc1[r] = ct[16 + half * 8 + r] * cv;
    acc2[r] = ct[32 + half * 8 + r] * cv;
    acc3[r] = ct[48 + half * 8 + r] * cv;
  }

#pragma unroll
  for (int kc = 0; kc < 16; ++kc) {
    // B fragment (4x16 f32): lanes 0-15 hold K=4kc..4kc+1, lanes 16-31 K=4kc+2..4kc+3
    v2f bf;
    bf.x = half ? zc[kc * 4 + 2] : zc[kc * 4 + 0];
    bf.y = half ? zc[kc * 4 + 3] : zc[kc * 4 + 1];
    int kbase = kc * 4 + half * 2;
    // A fragments (16x4 f32): lane holds A[M=lane%16][K=kbase..kbase+1]
    v2f a0; a0.x = Lt[(0 * 16 + col) * K_N + kbase]; a0.y = Lt[(0 * 16 + col) * K_N + kbase + 1];
    acc0 = __builtin_amdgcn_wmma_f32_16x16x4_f32(false, a0, false, bf, (short)0, acc0, false, false);
    v2f a1; a1.x = Lt[(1 * 16 + col) * K_N + kbase]; a1.y = Lt[(1 * 16 + col) * K_N + kbase + 1];
    acc1 = __builtin_amdgcn_wmma_f32_16x16x4_f32(false, a1, false, bf, (short)0, acc1, false, false);
    v2f a2; a2.x = Lt[(2 * 16 + col) * K_N + kbase]; a2.y = Lt[(2 * 16 + col) * K_N + kbase + 1];
    acc2 = __builtin_amdgcn_wmma_f32_16x16x4_f32(false, a2, false, bf, (short)0, acc2, false, false);
    v2f a3; a3.x = Lt[(3 * 16 + col) * K_N + kbase]; a3.y = Lt[(3 * 16 + col) * K_N + kbase + 1];
    acc3 = __builtin_amdgcn_wmma_f32_16x16x4_f32(false, a3, false, bf, (short)0, acc3, false, false);
  }

  // D layout: VGPR r, lane L -> M = r + 8*(L>=16), N = L%16
#pragma unroll
  for (int r = 0; r < 8; ++r) {
    int tp0 = half * 8 + r;
    size_t base = (size_t)s * T_N * 32 + (size_t)d * 8 + p;
    out[base + (size_t)(b * 64 + 0  + tp0) * 32] = acc0[r];
    out[base + (size_t)(b * 64 + 16 + tp0) * 32] = acc1[r];
    out[base + (size_t)(b * 64 + 32 + tp0) * 32] = acc2[r];
    out[base + (size_t)(b * 64 + 48 + tp0) * 32] = acc3[r];
  }
}

extern "C" void kernel_launch(void* const* d_in, const int* in_sizes, int n_in,
                              void* d_out, int out_size, void* d_ws, size_t ws_size,
                              hipStream_t stream) {
  (void)in_sizes; (void)n_in; (void)out_size; (void)ws_size;
  const float* m      = (const float*)d_in[2];
  const float* s_raw  = (const float*)d_in[3];
  const float* a_raw  = (const float*)d_in[4];
  const float* noise  = (const float*)d_in[5];
  const int*   dimidx = (const int*)d_in[7];
  float* out = (float*)d_out;
  float* ws  = (float*)d_ws;

  hipLaunchKernelGGL(kprep, dim3(128),  dim3(256), 0, stream, s_raw, a_raw, dimidx, ws);
  hipLaunchKernelGGL(kcoef, dim3(64),   dim3(64),  0, stream, ws);
  hipLaunchKernelGGL(kw,    dim3(512),  dim3(256), 0, stream, m, noise, dimidx, ws);
  hipLaunchKernelGGL(kscan, dim3(32),   dim3(256), 0, stream, ws);
  hipLaunchKernelGGL(kmain, dim3(1024), dim3(256), 0, stream, m, noise, dimidx, ws, out);
}